// WARPLoss_16879221473297
// MI455X (gfx1250) — compile-verified
//
#include <hip/hip_runtime.h>
#include <hip/hip_bf16.h>
#include <stdint.h>

// WARP loss: B=4096 rows, Y=10000 labels, T=128 trials.
// Roofline: 163.8 MB one-hot scan dominates (~7.3 us @ 23.3 TB/s); FLOPs
// negligible. No GEMM -> WMMA inapplicable. CDNA5 path: async global->LDS
// b128 copies (ASYNCcnt) double-buffered with s_wait_asynccnt, SADDR form
// + shared INST_OFFSET to minimize per-iteration address math, NT hint to
// keep the read-once stream out of L2.

constexpr int kB = 4096;
constexpr int kY = 10000;
constexpr int kT = 128;

// ---------------------------------------------------------------------------
// Kernel 1: stream target[B*Y] through LDS, find the one-hot column per row.
// Chunk = 256 floats per wave (2 x b128 async issues into one LDS buffer).
// Exactly one element per row is 1.0f -> plain store, no init, no atomics.
// ---------------------------------------------------------------------------
__launch_bounds__(256)
__global__ void warp_scan_target(const float* __restrict__ tgt,
                                 int* __restrict__ pos_idx,
                                 float* __restrict__ out) {
    if (blockIdx.x == 0 && threadIdx.x == 0) out[0] = 0.0f;

    __shared__ float4 buf[2][8][64];          // 16 KB: [parity][wave][2x32 lanes]

    const int lane = threadIdx.x & 31;
    const int wv   = threadIdx.x >> 5;

    const unsigned nChunks = (unsigned)((long)kB * kY / 256);  // 160000
    const unsigned wid     = blockIdx.x * 8u + (unsigned)wv;
    const unsigned stride  = gridDim.x * 8u;
    if (wid >= nChunks) return;               // wave-uniform

    const unsigned voff = (unsigned)lane * 16u;              // constant VGPR
    const unsigned lds0 = (unsigned)(uintptr_t)&buf[0][wv][lane];
    const unsigned lds1 = (unsigned)(uintptr_t)&buf[1][wv][lane];

    // Prologue: issue chunk `wid` into buffer 0 (2 x 512B per wave).
    {
        const float* g = tgt + (size_t)wid * 256;
        asm volatile("global_load_async_to_lds_b128 %0, %1, %2 th:TH_LOAD_NT"
                     :: "v"(lds0), "v"(voff), "s"(g) : "memory");
        asm volatile("global_load_async_to_lds_b128 %0, %1, %2 offset:512 th:TH_LOAD_NT"
                     :: "v"(lds0), "v"(voff), "s"(g) : "memory");
    }

    int parity = 0;
    for (unsigned c = wid; c < nChunks; c += stride) {
        const unsigned nx = c + stride;
        if (nx < nChunks) {
            const float* g = tgt + (size_t)nx * 256;
            const unsigned l = parity ? lds0 : lds1;  // next buffer
            asm volatile("global_load_async_to_lds_b128 %0, %1, %2 th:TH_LOAD_NT"
                         :: "v"(l), "v"(voff), "s"(g) : "memory");
            asm volatile("global_load_async_to_lds_b128 %0, %1, %2 offset:512 th:TH_LOAD_NT"
                         :: "v"(l), "v"(voff), "s"(g) : "memory");
            asm volatile("s_wait_asynccnt 2" ::: "memory");
        } else {
            asm volatile("s_wait_asynccnt 0" ::: "memory");
        }

        const float4 a = buf[parity][wv][lane];
        const float4 b = buf[parity][wv][lane + 32];

        const float m0 = fmaxf(fmaxf(a.x, a.y), fmaxf(a.z, a.w));
        const float m1 = fmaxf(fmaxf(b.x, b.y), fmaxf(b.z, b.w));
        if (fmaxf(m0, m1) > 0.5f) {           // one-hot hit: ~1e-4 rate, cold
            const float va[8] = {a.x, a.y, a.z, a.w, b.x, b.y, b.z, b.w};
            const unsigned base = c * 256u + (unsigned)lane * 4u;
            #pragma unroll
            for (int k = 0; k < 8; ++k) {
                if (va[k] > 0.5f) {
                    const unsigned gi  = base + (unsigned)(k < 4 ? k : 124 + k);
                    const unsigned row = gi / (unsigned)kY;
                    const unsigned col = gi - row * (unsigned)kY;
                    pos_idx[row] = (int)col;
                }
            }
        }
        parity ^= 1;
    }
}

// ---------------------------------------------------------------------------
// Kernel 2: one wave32 per row. Gather 128 candidate scores (4 per lane),
// find the first trial with margin >= 0 via a lane-min reduction, compute
// the WARP loss term and accumulate into out[0].
// ---------------------------------------------------------------------------
__launch_bounds__(256)
__global__ void warp_loss(const float* __restrict__ inp,
                          const int* __restrict__ neg,
                          const int* __restrict__ pos_idx,
                          float* __restrict__ out) {
    const int lane = threadIdx.x & 31;
    const int wv   = threadIdx.x >> 5;
    const int row  = blockIdx.x * 8 + wv;
    if (row >= kB) return;

    const float* rp = inp + (size_t)row * kY;
    const int*   nr = neg + (size_t)row * kT;
    const float  s_pos = rp[pos_idx[row]];

    int best = 0x7FFFFFFF;
    #pragma unroll
    for (int t = lane; t < kT; t += 32) {
        const float s = rp[nr[t]];
        if (1.0f + s - s_pos >= 0.0f) best = min(best, t);
    }
    #pragma unroll
    for (int off = 16; off > 0; off >>= 1)
        best = min(best, __shfl_xor(best, off, 32));

    if (lane == 0 && best < kT) {
        const int   num_trials = best + 1;
        const float s_neg = rp[nr[best]];
        const float L = logf((float)((kY - 1) / num_trials));
        atomicAdd(out, L * (1.0f - s_pos + s_neg));
    }
}

// ---------------------------------------------------------------------------
extern "C" void kernel_launch(void* const* d_in, const int* in_sizes, int n_in,
                              void* d_out, int out_size, void* d_ws, size_t ws_size,
                              hipStream_t stream) {
    const float* inp = (const float*)d_in[0];   // [B, Y] f32
    const float* tgt = (const float*)d_in[1];   // [B, Y] f32 one-hot
    const int*   neg = (const int*)d_in[2];     // [B, T] i32
    float*       out = (float*)d_out;           // [1] f32
    int*         pos = (int*)d_ws;              // B ints of scratch

    (void)in_sizes; (void)n_in; (void)out_size; (void)ws_size;

    // Kernel 1: 2048 blocks x 8 waves = 16384 waves over 160000 chunks.
    warp_scan_target<<<dim3(2048), dim3(256), 0, stream>>>(tgt, pos, out);

    // Kernel 2: one wave per row -> 4096/8 = 512 blocks.
    warp_loss<<<dim3(kB / 8), dim3(256), 0, stream>>>(inp, neg, pos, out);
}